// HVC_29781303231182
// MI455X (gfx1250) — compile-verified
//
#include <hip/hip_runtime.h>
#include <math.h>

// ---------------------------------------------------------------------------
// MI455X (gfx1250) implementation. fp32 end-to-end (memory-bound problem ->
// no reward for lower precision). Heavy GEMMs use V_WMMA_F32_16X16X4_F32.
// The dominant Q^T K kernel stages 64x64 tiles into LDS with
// GLOBAL_LOAD_ASYNC_TO_LDS_B128 (ASYNCcnt) and feeds WMMA from LDS.
// ---------------------------------------------------------------------------

typedef float v2f __attribute__((ext_vector_type(2)));
typedef float v8f __attribute__((ext_vector_type(8)));

#define NB    32
#define CCH   256
#define HW    1024
#define NPIX  (NB * HW)                 // 32768
#define TENS  ((size_t)NB * CCH * HW)   // 8388608 floats per [N,C,H,W] tensor

// -------------------- l2 normalize over channel dim ------------------------
__global__ void hvc_l2norm(const float* __restrict__ in, float* __restrict__ out) {
  const int pix0 = blockIdx.x * 32;
  const int n  = pix0 >> 10;
  const int s0 = pix0 & (HW - 1);
  const int sx = threadIdx.x, cy = threadIdx.y;
  const float* bi = in + (size_t)n * CCH * HW + s0 + sx;
  float v[32];
  float ssq = 0.f;
#pragma unroll
  for (int i = 0; i < 32; ++i) {
    float x = bi[(size_t)(cy + i * 8) * HW];
    v[i] = x; ssq += x * x;
  }
  __shared__ float red[8][32];
  red[cy][sx] = ssq;
  __syncthreads();
  float tot = 0.f;
#pragma unroll
  for (int j = 0; j < 8; ++j) tot += red[j][sx];
  const float inv = 1.0f / fmaxf(sqrtf(tot), 1e-12f);
  float* bo = out + (size_t)n * CCH * HW + s0 + sx;
#pragma unroll
  for (int i = 0; i < 32; ++i) bo[(size_t)(cy + i * 8) * HW] = v[i] * inv;
}

// -------------------- conv1x1 #1 as WMMA GEMM ------------------------------
// hidden[o,pix] = sum_c W1[o,c] * concat(xa,xb)[c,pix] + b1[o]
__global__ void __launch_bounds__(32)
hvc_conv1_wmma(const float* __restrict__ xa, const float* __restrict__ xb,
               const float* __restrict__ w1, const float* __restrict__ b1,
               float* __restrict__ hidden) {
  const int pix0 = blockIdx.x * 16;
  const int o0   = blockIdx.y * 16;
  const int lane = threadIdx.x;
  const int m    = lane & 15;
  const int kg   = lane >> 4;
  const int n    = pix0 >> 10;
  const int s    = (pix0 & (HW - 1)) + m;
  const float* wrow = w1 + (size_t)(o0 + m) * 512;
  const float* pa = xa + (size_t)n * CCH * HW + s;
  const float* pb = xb + (size_t)n * CCH * HW + s;
  __builtin_prefetch(pa, 0, 3);
  __builtin_prefetch(pb, 0, 3);
  v8f acc = {};
  for (int kk = 0; kk < 256; kk += 4) {
    const int c0 = kk + 2 * kg;
    v2f af, bf;
    af.x = wrow[c0];               af.y = wrow[c0 + 1];
    bf.x = pa[(size_t)c0 * HW];    bf.y = pa[(size_t)(c0 + 1) * HW];
    acc = __builtin_amdgcn_wmma_f32_16x16x4_f32(false, af, false, bf,
                                                (short)0, acc, false, false);
  }
  for (int kk = 0; kk < 256; kk += 4) {
    const int c0 = kk + 2 * kg;
    v2f af, bf;
    af.x = wrow[256 + c0];         af.y = wrow[256 + c0 + 1];
    bf.x = pb[(size_t)c0 * HW];    bf.y = pb[(size_t)(c0 + 1) * HW];
    acc = __builtin_amdgcn_wmma_f32_16x16x4_f32(false, af, false, bf,
                                                (short)0, acc, false, false);
  }
  const int pix = pix0 + m;
#pragma unroll
  for (int r = 0; r < 8; ++r) {
    const int o = o0 + r + 8 * kg;
    hidden[(size_t)o * NPIX + pix] = acc[r] + b1[o];
  }
}

// -------------------- BN1 statistics (per channel, biased var) -------------
__global__ void hvc_bnstats(const float* __restrict__ hidden,
                            float* __restrict__ mean, float* __restrict__ var) {
  const int ch = blockIdx.x;
  const int t  = threadIdx.x;  // 256 threads
  const float* row = hidden + (size_t)ch * NPIX;
  float s = 0.f, q = 0.f;
  for (int i = t; i < NPIX; i += 256) { float v = row[i]; s += v; q += v * v; }
  __shared__ float ss[256], sq[256];
  ss[t] = s; sq[t] = q;
  __syncthreads();
  for (int o = 128; o > 0; o >>= 1) {
    if (t < o) { ss[t] += ss[t + o]; sq[t] += sq[t + o]; }
    __syncthreads();
  }
  if (t == 0) {
    const float mu = ss[0] * (1.0f / NPIX);
    mean[ch] = mu;
    var[ch]  = sq[0] * (1.0f / NPIX) - mu * mu;
  }
}

// -------------------- BN1 + ReLU + conv1x1 #2 fused ------------------------
__global__ void hvc_conv2(const float* __restrict__ hidden,
                          const float* __restrict__ mean, const float* __restrict__ var,
                          const float* __restrict__ g1, const float* __restrict__ be1,
                          const float* __restrict__ w2, const float* __restrict__ b2,
                          float* __restrict__ ytmp, float* __restrict__ ystats) {
  const int sx = threadIdx.x, cy = threadIdx.y;
  const int pix = blockIdx.x * 32 + sx;
  float y0 = 0.f, y1 = 0.f;
  for (int c = cy; c < 256; c += 8) {
    const float h  = hidden[(size_t)c * NPIX + pix];
    float hn = (h - mean[c]) * rsqrtf(var[c] + 1e-5f) * g1[c] + be1[c];
    hn = fmaxf(hn, 0.f);
    y0 += w2[c] * hn;
    y1 += w2[256 + c] * hn;
  }
  __shared__ float r0[8][32], r1[8][32];
  r0[cy][sx] = y0; r1[cy][sx] = y1;
  __syncthreads();
  if (cy == 0) {                       // threads 0..31 == one full wave32
    float a = b2[0], b = b2[1];
#pragma unroll
    for (int j = 0; j < 8; ++j) { a += r0[j][sx]; b += r1[j][sx]; }
    ytmp[pix] = a; ytmp[NPIX + pix] = b;
    float s0 = a, q0 = a * a, s1 = b, q1 = b * b;
    for (int o = 16; o > 0; o >>= 1) {
      s0 += __shfl_down(s0, o);
      q0 += __shfl_down(q0, o);
      s1 += __shfl_down(s1, o);
      q1 += __shfl_down(q1, o);
    }
    if (sx == 0) {
      atomicAdd(&ystats[0], s0); atomicAdd(&ystats[1], q0);
      atomicAdd(&ystats[2], s1); atomicAdd(&ystats[3], q1);
    }
  }
}

// -------------------- BN2 + l2norm over the 2 flow channels ----------------
__global__ void hvc_bn2_l2(const float* __restrict__ ytmp, const float* __restrict__ ystats,
                           const float* __restrict__ g2, const float* __restrict__ be2,
                           float* __restrict__ flow) {
  const int pix = blockIdx.x * 256 + threadIdx.x;
  const float invn = 1.0f / NPIX;
  const float m0 = ystats[0] * invn, v0 = ystats[1] * invn - m0 * m0;
  const float m1 = ystats[2] * invn, v1 = ystats[3] * invn - m1 * m1;
  const float z0 = (ytmp[pix]        - m0) * rsqrtf(v0 + 1e-5f) * g2[0] + be2[0];
  const float z1 = (ytmp[NPIX + pix] - m1) * rsqrtf(v1 + 1e-5f) * g2[1] + be2[1];
  const float inv = 1.0f / fmaxf(sqrtf(z0 * z0 + z1 * z1), 1e-12f);
  flow[pix]        = z0 * inv;
  flow[NPIX + pix] = z1 * inv;
}

// -------------------- fused Q^T K WMMA + mask + temporal reduction ---------
// 4-wave workgroup computes a 64(p) x 64(q) tile per batch.  Each 32-channel
// chunk of Q/K is staged into LDS with GLOBAL_LOAD_ASYNC_TO_LDS_B128
// (tracked by ASYNCcnt, drained with S_WAIT_ASYNCCNT), then all waves feed
// V_WMMA_F32_16X16X4_F32 from LDS.  Wave w owns the 16-row strip p0+16w and
// 4 accumulators covering the 64 q-columns (A-fragment reused 4x).
__global__ void __launch_bounds__(128)
hvc_logit_mask(const float* __restrict__ Q, const float* __restrict__ K,
               const float* __restrict__ ff, const float* __restrict__ fb,
               const float* __restrict__ cq, const float* __restrict__ ck,
               float* __restrict__ acc) {
  __shared__ float Qs[32][64];   // 8 KB: [chunk-channel][p]
  __shared__ float Ks[32][64];   // 8 KB: [chunk-channel][q]
  const int n    = blockIdx.y;
  const int p0   = (blockIdx.x >> 4) * 64;
  const int q0   = (blockIdx.x & 15) * 64;
  const int tid  = threadIdx.x;      // 0..127
  const int wave = tid >> 5;         // 0..3
  const int lane = tid & 31;
  const int m    = lane & 15;
  const int kg   = lane >> 4;
  const float* qbase = Q + (size_t)n * CCH * HW;
  const float* kbase = K + (size_t)n * CCH * HW;
  const unsigned qs_off = (unsigned)(size_t)&Qs[0][0];  // LDS byte offset
  const unsigned ks_off = (unsigned)(size_t)&Ks[0][0];

  v8f accv[4] = {};
  for (int cb = 0; cb < CCH; cb += 32) {
    // ---- async stage: 32ch x 64px x 4B per matrix = 512 x b128 lanes ----
#pragma unroll
    for (int i = 0; i < 4; ++i) {
      const int idx  = i * 128 + tid;        // 0..511
      const int row  = idx >> 4;             // chunk channel 0..31
      const int col4 = (idx & 15) << 2;      // 0,4,...,60
      const unsigned lq = qs_off + (unsigned)((row * 64 + col4) * 4);
      const unsigned lk = ks_off + (unsigned)((row * 64 + col4) * 4);
      const float* gq = qbase + (size_t)(cb + row) * HW + p0 + col4;
      const float* gk = kbase + (size_t)(cb + row) * HW + q0 + col4;
      asm volatile("global_load_async_to_lds_b128 %0, %1, off"
                   :: "v"(lq), "v"(gq) : "memory");
      asm volatile("global_load_async_to_lds_b128 %0, %1, off"
                   :: "v"(lk), "v"(gk) : "memory");
    }
    asm volatile("s_wait_asynccnt 0x0" ::: "memory");
    __syncthreads();
    // ---- WMMA from LDS: 8 k-steps x 4 q-subtiles ----
    const int pr = wave * 16 + m;            // A fragment row (M = lane&15)
#pragma unroll
    for (int c = 0; c < 32; c += 4) {
      const int c0 = c + 2 * kg;
      v2f af;
      af.x = Qs[c0][pr];
      af.y = Qs[c0 + 1][pr];
#pragma unroll
      for (int sq = 0; sq < 4; ++sq) {
        v2f bf;
        bf.x = Ks[c0][sq * 16 + m];
        bf.y = Ks[c0 + 1][sq * 16 + m];
        accv[sq] = __builtin_amdgcn_wmma_f32_16x16x4_f32(false, af, false, bf,
                                                         (short)0, accv[sq],
                                                         false, false);
      }
    }
    __syncthreads();                         // protect LDS before next chunk
  }

  // ---- geometric mask scalars for this batch ----
  const float qx0 = cq[n * 4 + 0], qy0 = cq[n * 4 + 1];
  const float qx1 = cq[n * 4 + 2], qy1 = cq[n * 4 + 3];
  const float kx0 = ck[n * 4 + 0], ky0 = ck[n * 4 + 1];
  const float kx1 = ck[n * 4 + 2], ky1 = ck[n * 4 + 3];
  const float qbw = (qx1 - qx0) * (1.f / 32.f), qbh = (qy1 - qy0) * (1.f / 32.f);
  const float kbw = (kx1 - kx0) * (1.f / 32.f), kbh = (ky1 - ky0) * (1.f / 32.f);
  const float md = fmaxf(sqrtf(qbw * qbw + qbh * qbh), sqrtf(kbw * kbw + kbh * kbh));
  const float r2 = 0.25f * md * md;          // (0.5 * max_diag)^2

  // this wave's 8 p rows (fixed across subtiles)
  float cqxv[8], cqyv[8], f0v[8], f1v[8];
#pragma unroll
  for (int r = 0; r < 8; ++r) {
    const int p = p0 + wave * 16 + r + 8 * kg;
    cqxv[r] = ((float)(p & 31) + 0.5f) * qbw + qx0;
    cqyv[r] = ((float)(p >> 5) + 0.5f) * qbh + qy0;
    f0v[r]  = ff[(size_t)n * HW + p];
    f1v[r]  = ff[(size_t)NPIX + n * HW + p];
  }

  float sp = 0.f, tp = 0.f, ms = 0.f;
#pragma unroll
  for (int sq = 0; sq < 4; ++sq) {
    const int   qq  = q0 + sq * 16 + m;
    const float ckx = ((float)(qq & 31) + 0.5f) * kbw + kx0;
    const float cky = ((float)(qq >> 5) + 0.5f) * kbh + ky0;
    const float gb0 = fb[(size_t)n * HW + qq];
    const float gb1 = fb[(size_t)NPIX + n * HW + qq];
#pragma unroll
    for (int r = 0; r < 8; ++r) {
      const float dx = cqxv[r] - ckx, dy = cqyv[r] - cky;
      const float mk = (dx * dx + dy * dy < r2) ? 1.0f : 0.0f;
      sp += mk * accv[sq][r];
      tp += mk * (f0v[r] * gb0 + f1v[r] * gb1);
      ms += mk;
    }
  }
  for (int o = 16; o > 0; o >>= 1) {
    sp += __shfl_xor(sp, o);
    tp += __shfl_xor(tp, o);
    ms += __shfl_xor(ms, o);
  }
  if (lane == 0) {
    atomicAdd(&acc[n],      sp);
    atomicAdd(&acc[32 + n], tp);
    atomicAdd(&acc[64 + n], ms);
  }
}

// -------------------- final scalar loss ------------------------------------
__global__ void hvc_final(const float* __restrict__ acc, float* __restrict__ out) {
  const int t = threadIdx.x;                 // 64 = 2 calls x 32 batches
  const float* a = acc + (t >> 5) * 96;
  const int n = t & 31;
  const float ms = a[64 + n] + 1e-6f;
  float v = (a[n] + a[32 + n]) / ms;
  __shared__ float red[64];
  red[t] = v;
  __syncthreads();
  for (int o = 32; o > 0; o >>= 1) {
    if (t < o) red[t] += red[t + o];
    __syncthreads();
  }
  if (t == 0) out[0] = red[0] * (-2.0f / 32.0f);
}

__global__ void hvc_zero(float* __restrict__ p, int count) {
  const int i = blockIdx.x * 256 + threadIdx.x;
  if (i < count) p[i] = 0.f;
}

// ---------------------------------------------------------------------------
extern "C" void kernel_launch(void* const* d_in, const int* in_sizes, int n_in,
                              void* d_out, int out_size, void* d_ws, size_t ws_size,
                              hipStream_t stream) {
  (void)in_sizes; (void)n_in; (void)out_size; (void)ws_size;
  const float* q1 = (const float*)d_in[0];
  const float* k1 = (const float*)d_in[1];
  const float* q2 = (const float*)d_in[2];
  const float* k2 = (const float*)d_in[3];
  const float* coord1 = (const float*)d_in[4];
  const float* coord2 = (const float*)d_in[5];
  const float* w1  = (const float*)d_in[6];
  const float* b1  = (const float*)d_in[7];
  const float* g1  = (const float*)d_in[8];
  const float* be1 = (const float*)d_in[9];
  const float* w2  = (const float*)d_in[10];
  const float* b2  = (const float*)d_in[11];
  const float* g2  = (const float*)d_in[12];
  const float* be2 = (const float*)d_in[13];

  float* ws  = (float*)d_ws;
  float* NQ1 = ws;
  float* NK1 = ws + TENS;
  float* NQ2 = ws + 2 * TENS;
  float* NK2 = ws + 3 * TENS;
  float* HID = ws + 4 * TENS;                      // 256 x 32768
  float* FLW = ws + 5 * TENS;                      // 4 flow maps, 2 x 32768 each
  float* YT  = FLW + (size_t)4 * 2 * NPIX;         // 2 x 32768
  float* BNM = YT + (size_t)2 * NPIX;              // 256
  float* BNV = BNM + 256;                          // 256
  float* ACC = BNV + 256;                          // 2 calls x 96
  float* YST = ACC + 192;                          // 4 flows x 4 stats

  hvc_zero<<<1, 256, 0, stream>>>(ACC, 208);

  const dim3 bl(32, 8);
  hvc_l2norm<<<NPIX / 32, bl, 0, stream>>>(q1, NQ1);
  hvc_l2norm<<<NPIX / 32, bl, 0, stream>>>(k1, NK1);
  hvc_l2norm<<<NPIX / 32, bl, 0, stream>>>(q2, NQ2);
  hvc_l2norm<<<NPIX / 32, bl, 0, stream>>>(k2, NK2);

  // flow evaluations: call0: f=flow(nq1,nk2), b=flow(nk2,nq1)
  //                   call1: f=flow(nq2,nk1), b=flow(nk1,nq2)
  const float* fa[4]  = {NQ1, NK2, NQ2, NK1};
  const float* fbp[4] = {NK2, NQ1, NK1, NQ2};
  for (int i = 0; i < 4; ++i) {
    hvc_conv1_wmma<<<dim3(NPIX / 16, 16), 32, 0, stream>>>(fa[i], fbp[i], w1, b1, HID);
    hvc_bnstats<<<256, 256, 0, stream>>>(HID, BNM, BNV);
    hvc_conv2<<<NPIX / 32, bl, 0, stream>>>(HID, BNM, BNV, g1, be1, w2, b2, YT, YST + 4 * i);
    hvc_bn2_l2<<<NPIX / 256, 256, 0, stream>>>(YT, YST + 4 * i, g2, be2,
                                               FLW + (size_t)i * 2 * NPIX);
  }

  hvc_logit_mask<<<dim3(256, NB), 128, 0, stream>>>(NQ1, NK2,
      FLW + 0,                 FLW + (size_t)2 * NPIX, coord1, coord2, ACC);
  hvc_logit_mask<<<dim3(256, NB), 128, 0, stream>>>(NQ2, NK1,
      FLW + (size_t)4 * NPIX,  FLW + (size_t)6 * NPIX, coord2, coord1, ACC + 96);

  hvc_final<<<1, 64, 0, stream>>>(ACC, (float*)d_out);
}